// DetectionLoss_58755152609322
// MI455X (gfx1250) — compile-verified
//
#include <hip/hip_runtime.h>
#include <hip/hip_bf16.h>
#include <math.h>

#define POS_T     0.5f
#define RESCUE_T  0.3f
#define ALPHA_C   0.6f
#define EPS_F     1e-6f
#define TGT       32
#define MAXPOSW   3200   // bitmask words -> supports A <= 102400

typedef __attribute__((ext_vector_type(2))) float v2f;
typedef __attribute__((ext_vector_type(8))) float v8f;

__device__ __forceinline__ float box_area(float4 b) {
    return (b.z - b.x) * (b.w - b.y);
}

__device__ __forceinline__ float iou_pair(float4 a, float aa, float4 t, float tat) {
    float ix = fminf(a.z, t.z) - fmaxf(a.x, t.x);
    float iy = fminf(a.w, t.w) - fmaxf(a.y, t.y);
    ix = fmaxf(ix, 0.f); iy = fmaxf(iy, 0.f);
    float inter = ix * iy;
    return inter / (aa + tat - inter + EPS_F);
}

// ---------------- Phase 1: per-anchor row max / argmax over targets ----------
__global__ void k_rowmax(const float4* __restrict__ anchors,
                         const float4* __restrict__ tboxes,
                         float* __restrict__ maxiou, int* __restrict__ bestt, int A) {
    __shared__ float4 tb[TGT];
    __shared__ float  ta[TGT];
    const int b = blockIdx.y, tid = threadIdx.x;
    if (tid < TGT) { float4 t = tboxes[b * TGT + tid]; tb[tid] = t; ta[tid] = box_area(t); }
    __syncthreads();
    int a = blockIdx.x * blockDim.x + tid;
    if (a >= A) return;
    float4 an = anchors[a];
    float  aa = box_area(an);
    float best = -1.f; int bi = 0;
    #pragma unroll
    for (int t = 0; t < TGT; ++t) {
        float iou = iou_pair(an, aa, tb[t], ta[t]);
        if (iou > best) { best = iou; bi = t; }   // first occurrence wins ties
    }
    size_t idx = (size_t)b * A + a;
    maxiou[idx] = best;
    bestt[idx]  = bi;
}

// ---------------- Phase 2: per-target argmax over anchors (best_a) -----------
__global__ void k_besta(const float4* __restrict__ anchors,
                        const float4* __restrict__ tboxes,
                        int* __restrict__ besta, int A) {
    const int b = blockIdx.y, t = blockIdx.x, tid = threadIdx.x;
    __shared__ float sv[256];
    __shared__ int   si[256];
    float4 tbx = tboxes[b * TGT + t];
    float  tat = box_area(tbx);
    float v = -1.f; int vi = 0x7FFFFFFF;
    for (int a = tid; a < A; a += 256) {
        float4 an = anchors[a];
        float iou = iou_pair(an, box_area(an), tbx, tat);
        if (iou > v) { v = iou; vi = a; }         // ascending a => min index on ties
    }
    sv[tid] = v; si[tid] = vi;
    __syncthreads();
    for (int s = 128; s > 0; s >>= 1) {
        if (tid < s) {
            float w = sv[tid + s]; int wi = si[tid + s];
            if (w > sv[tid] || (w == sv[tid] && wi < si[tid])) { sv[tid] = w; si[tid] = wi; }
        }
        __syncthreads();
    }
    if (tid == 0) besta[b * TGT + t] = si[0];
}

// ---------------- Phase 3: per-image assignment scan + loss ------------------
__global__ __launch_bounds__(1024)
void k_image(const float4* __restrict__ anchors,
             const float4* __restrict__ tboxes,
             const float4* __restrict__ bboxp,
             const float*  __restrict__ confp,
             const float*  __restrict__ maxiou,
             const int*    __restrict__ bestt,
             const int*    __restrict__ besta,
             float* __restrict__ imgres, int A) {
    const int b = blockIdx.x, tid = threadIdx.x, NT = blockDim.x;
    const size_t base = (size_t)b * A;
    __shared__ float4   tb[TGT];
    __shared__ float    ta[TGT];
    __shared__ unsigned posm[MAXPOSW];
    __shared__ unsigned mflag[TGT];
    __shared__ float t2v0[1024]; __shared__ int t2i0[1024];
    __shared__ float t2v1[1024]; __shared__ int t2i1[1024];
    __shared__ float rx[1024], ry[1024], rz[1024], rw[1024];

    if (tid < TGT) {
        float4 t = tboxes[b * TGT + tid];
        tb[tid] = t; ta[tid] = box_area(t); mflag[tid] = 0u;
    }
    // pos0 bitmask: each thread owns whole words -> no atomics needed
    const int nw = (A + 31) >> 5;
    for (int w = tid; w < nw; w += NT) {
        unsigned bits = 0u;
        int a0 = w << 5;
        int lim = (A - a0 < 32) ? (A - a0) : 32;
        for (int j = 0; j < lim; ++j)
            if (maxiou[base + a0 + j] >= POS_T) bits |= (1u << j);
        posm[w] = bits;
    }
    __syncthreads();
    // force best_a anchors positive
    if (tid < TGT) {
        int a = besta[b * TGT + tid];
        atomicOr(&posm[a >> 5], 1u << (a & 31));
    }
    __syncthreads();
    // matched histogram: mflag[t] = any(pos & best_t==t)
    for (int a = tid; a < A; a += NT) {
        if ((posm[a >> 5] >> (a & 31)) & 1u) mflag[bestt[base + a]] = 1u;
    }
    __syncthreads();

    // sequential rescue scan (top-2 pass only for unmatched targets)
    for (int t = 0; t < TGT; ++t) {
        if (mflag[t] == 0u) {   // uniform branch (shared value)
            float4 tbx = tb[t]; float tat = ta[t];
            float v0 = -2.f, v1 = -2.f; int i0 = 0x7FFFFFFF, i1 = 0x7FFFFFFF;
            for (int a = tid; a < A; a += NT) {
                if ((posm[a >> 5] >> (a & 31)) & 1u) continue;
                float4 an = anchors[a];
                float iou = iou_pair(an, box_area(an), tbx, tat);
                if (iou >= RESCUE_T) {
                    if (iou > v1 || (iou == v1 && a < i1)) {
                        if (iou > v0 || (iou == v0 && a < i0)) { v1 = v0; i1 = i0; v0 = iou; i0 = a; }
                        else                                   { v1 = iou; i1 = a; }
                    }
                }
            }
            t2v0[tid] = v0; t2i0[tid] = i0; t2v1[tid] = v1; t2i1[tid] = i1;
            __syncthreads();
            for (int s = NT >> 1; s > 0; s >>= 1) {
                if (tid < s) {
                    float w0 = t2v0[tid + s]; int j0 = t2i0[tid + s];
                    float w1 = t2v1[tid + s]; int j1 = t2i1[tid + s];
                    float p0 = t2v0[tid];     int q0 = t2i0[tid];
                    float p1 = t2v1[tid];     int q1 = t2i1[tid];
                    if (w0 > p0 || (w0 == p0 && j0 < q0)) {
                        float sv2; int si2;
                        if (w1 > p0 || (w1 == p0 && j1 < q0)) { sv2 = w1; si2 = j1; }
                        else                                   { sv2 = p0; si2 = q0; }
                        p0 = w0; q0 = j0; p1 = sv2; q1 = si2;
                    } else {
                        if (w0 > p1 || (w0 == p1 && j0 < q1)) { p1 = w0; q1 = j0; }
                    }
                    t2v0[tid] = p0; t2i0[tid] = q0; t2v1[tid] = p1; t2i1[tid] = q1;
                }
                __syncthreads();
            }
            if (tid == 0) {
                float f0 = t2v0[0]; int g0 = t2i0[0];
                float f1 = t2v1[0]; int g1 = t2i1[0];
                if (f0 >= RESCUE_T) { posm[g0 >> 5] |= (1u << (g0 & 31)); mflag[bestt[base + g0]] = 1u; }
                if (f1 >= RESCUE_T) { posm[g1 >> 5] |= (1u << (g1 & 31)); mflag[bestt[base + g1]] = 1u; }
            }
            __syncthreads();
        }
    }

    // loss pass: focal BCE over all anchors; conf-weighted DIoU over positives
    float s_bw = 0.f, s_w = 0.f, s_dl = 0.f, s_np = 0.f;
    for (int a = tid; a < A; a += NT) {
        float conf = confp[base + a];
        bool  p = (posm[a >> 5] >> (a & 31)) & 1u;
        float mi = maxiou[base + a];
        float u = p ? (1.f - conf) : conf;                 // (1 - pt)
        float alpha_f = p ? (1.f - ALPHA_C) : ALPHA_C;
        float iou_w = p ? mi * mi : 1.f;
        float wgt = u * sqrtf(u) * alpha_f * iou_w;        // (1-pt)^1.5 * alpha * iou_w
        float lg = p ? logf(conf) : logf(1.f - conf);
        lg = fmaxf(lg, -100.f);
        s_bw += (-lg) * wgt;
        s_w  += wgt;
        if (p) {
            int t = bestt[base + a];
            float4 b1 = bboxp[base + a];
            float4 b2 = tb[t];
            float ix = fminf(b1.z, b2.z) - fmaxf(b1.x, b2.x);
            float iy = fminf(b1.w, b2.w) - fmaxf(b1.y, b2.y);
            ix = fmaxf(ix, 0.f); iy = fmaxf(iy, 0.f);
            float inter = ix * iy;
            float a1 = box_area(b1), a2 = ta[t];
            float iou = inter / (a1 + a2 - inter + EPS_F);
            float c1x = (b1.x + b1.z) * 0.5f, c1y = (b1.y + b1.w) * 0.5f;
            float c2x = (b2.x + b2.z) * 0.5f, c2y = (b2.y + b2.w) * 0.5f;
            float dx = c1x - c2x, dy = c1y - c2y;
            float cd2 = dx * dx + dy * dy;
            float ex = fmaxf(b1.z, b2.z) - fminf(b1.x, b2.x);
            float ey = fmaxf(b1.w, b2.w) - fminf(b1.y, b2.y);
            float diag2 = ex * ex + ey * ey;
            float dl = 1.f - (iou - cd2 / (diag2 + EPS_F));
            s_dl += dl * conf;
            s_np += 1.f;
        }
    }
    rx[tid] = s_bw; ry[tid] = s_w; rz[tid] = s_dl; rw[tid] = s_np;
    __syncthreads();
    for (int s = NT >> 1; s > 0; s >>= 1) {
        if (tid < s) {
            rx[tid] += rx[tid + s]; ry[tid] += ry[tid + s];
            rz[tid] += rz[tid + s]; rw[tid] += rw[tid + s];
        }
        __syncthreads();
    }
    if (tid == 0) {
        imgres[2 * b]     = rx[0] / (ry[0] + EPS_F);
        imgres[2 * b + 1] = rz[0] / fmaxf(rw[0], 1.f);
    }
}

// ---------------- Phase 4: combine per-image losses via WMMA -----------------
__global__ void k_final(const float* __restrict__ imgres, float* __restrict__ out, int B) {
    if (B == 16) {
        int lane = threadIdx.x;              // 32 threads, EXEC all ones
        int n = lane & 15;
        float v = imgres[2 * n + (lane >> 4)];   // lanes 0-15: conf_l, 16-31: bbox_l
#if __has_builtin(__builtin_amdgcn_wmma_f32_16x16x4_f32)
        v2f av = {1.f, 1.f};                 // A = ones(16x4)
        v2f bv = {v, 0.f};                   // B rows 0/2 = data, rows 1/3 = 0
        v8f c = {};
        c = __builtin_amdgcn_wmma_f32_16x16x4_f32(false, av, false, bv,
                                                  (short)0, c, false, false);
        // sum entire 16x16 D tile = 16 * (sum conf_l + sum bbox_l), layout-invariant
        float s = 0.f;
        #pragma unroll
        for (int i = 0; i < 8; ++i) s += c[i];
        #pragma unroll
        for (int off = 16; off > 0; off >>= 1) s += __shfl_xor(s, off, 32);
        if (lane == 0) out[0] = s * (1.f / 256.f);   // (total)/16 = mean+mean
#else
        float s = v;
        #pragma unroll
        for (int off = 16; off > 0; off >>= 1) s += __shfl_xor(s, off, 32);
        if (lane == 0) out[0] = s * (1.f / 16.f);
#endif
    } else {
        if (threadIdx.x == 0) {
            float cs = 0.f, bs = 0.f;
            for (int i = 0; i < B; ++i) { cs += imgres[2 * i]; bs += imgres[2 * i + 1]; }
            out[0] = cs / (float)B + bs / (float)B;
        }
    }
}

static inline size_t align256(size_t x) { return (x + 255) & ~(size_t)255; }

extern "C" void kernel_launch(void* const* d_in, const int* in_sizes, int n_in,
                              void* d_out, int out_size, void* d_ws, size_t ws_size,
                              hipStream_t stream) {
    const float* bbox_pred = (const float*)d_in[0];   // [B,A,4]
    const float* conf_pred = (const float*)d_in[1];   // [B,A]
    const float* anchors   = (const float*)d_in[2];   // [A,4]
    const float* tboxes    = (const float*)d_in[3];   // [B,T,4]
    (void)n_in; (void)out_size; (void)ws_size;

    const int A = in_sizes[2] / 4;
    const int B = in_sizes[1] / A;

    char* w = (char*)d_ws;
    size_t o = 0;
    float* maxiou = (float*)(w + o); o += align256((size_t)B * A * sizeof(float));
    int*   bestt  = (int*)  (w + o); o += align256((size_t)B * A * sizeof(int));
    int*   besta  = (int*)  (w + o); o += align256((size_t)B * TGT * sizeof(int));
    float* imgres = (float*)(w + o);

    const float4* an4 = (const float4*)anchors;
    const float4* tb4 = (const float4*)tboxes;
    const float4* bp4 = (const float4*)bbox_pred;

    dim3 g1((A + 255) / 256, B);
    k_rowmax<<<g1, 256, 0, stream>>>(an4, tb4, maxiou, bestt, A);

    dim3 g2(TGT, B);
    k_besta<<<g2, 256, 0, stream>>>(an4, tb4, besta, A);

    k_image<<<B, 1024, 0, stream>>>(an4, tb4, bp4, conf_pred, maxiou, bestt, besta,
                                    imgres, A);

    k_final<<<1, 32, 0, stream>>>(imgres, (float*)d_out, B);
}